// MemSACLoss_69406671503848
// MI455X (gfx1250) — compile-verified
//
#include <hip/hip_runtime.h>
#include <math.h>

#define DIM      256
#define QUEUE_N  48000
#define NCLS     126
#define BS       512
#define KTOP     5
#define WARM_UP  4000

// LDS B-tile row stride (floats): 256 + 4 pad -> bank stride 4, so the 32
// lanes of a ds_load_b64 B-fragment read cover all 64 banks conflict-free.
#define BT_STRIDE 260

typedef __attribute__((ext_vector_type(2))) float v2f;
typedef __attribute__((ext_vector_type(4))) float v4f;
typedef __attribute__((ext_vector_type(8))) float v8f;

// GCC-style vector to match the builtin's parameter type exactly
// ("int __vector(4)" in the clang diagnostic).
typedef int v4i_vs __attribute__((vector_size(16)));
typedef __attribute__((address_space(1))) v4i_vs as1_v4i;
typedef __attribute__((address_space(3))) v4i_vs as3_v4i;

#if defined(__has_builtin)
#if __has_builtin(__builtin_amdgcn_global_load_async_to_lds_b128) && \
    __has_builtin(__builtin_amdgcn_s_wait_asynccnt)
#define USE_ASYNC_LDS 1
#endif
#endif

__device__ __forceinline__ void copy16_g2lds(const float* g, float* l) {
#if defined(USE_ASYNC_LDS)
    __builtin_amdgcn_global_load_async_to_lds_b128(
        (as1_v4i*)(uintptr_t)g,
        (as3_v4i*)(uint32_t)(uintptr_t)l,
        0, 0);
#else
    *(v4f*)l = *(const v4f*)g;
#endif
}

__device__ __forceinline__ void wait_async_copies() {
#if defined(USE_ASYNC_LDS)
    __builtin_amdgcn_s_wait_asynccnt(0);
#endif
}

// ---------------------------------------------------------------------------
// Phase 0: build normalized A (tgt, 512x256), normalized B (new_queue,
// 48000x256) and merged labels. One wave (32 lanes) per row.
// ---------------------------------------------------------------------------
__global__ void __launch_bounds__(256)
prep_kernel(const float* __restrict__ feats, const int* __restrict__ src_lab,
            const float* __restrict__ queue, const int* __restrict__ q_lab,
            float* __restrict__ An, float* __restrict__ Bn,
            int* __restrict__ labels) {
    int gid  = blockIdx.x * blockDim.x + threadIdx.x;
    int wave = gid >> 5;
    int lane = threadIdx.x & 31;
    const int total = QUEUE_N + BS;
    if (wave >= total) return;

    const float* srcRow;
    float* dstRow;
    if (wave < QUEUE_N) {
        srcRow = (wave < BS) ? (feats + (size_t)wave * DIM)
                             : (queue + (size_t)wave * DIM);
        dstRow = Bn + (size_t)wave * DIM;
        if (lane == 0) labels[wave] = (wave < BS) ? src_lab[wave] : q_lab[wave];
    } else {
        int r  = wave - QUEUE_N;                 // tgt row
        srcRow = feats + (size_t)(BS + r) * DIM;
        dstRow = An + (size_t)r * DIM;
    }

    float vals[8];
    float ss = 0.f;
#pragma unroll
    for (int i = 0; i < 8; ++i) {
        vals[i] = srcRow[lane + 32 * i];
        ss += vals[i] * vals[i];
    }
#pragma unroll
    for (int off = 16; off >= 1; off >>= 1)
        ss += __shfl_xor(ss, off, 32);

    float inv = 1.0f / fmaxf(sqrtf(ss), 1e-12f);
#pragma unroll
    for (int i = 0; i < 8; ++i)
        dstRow[lane + 32 * i] = vals[i] * inv;
}

// ---------------------------------------------------------------------------
// Phase 1: sim = An (512x256) x Bn^T (256x48000) with V_WMMA_F32_16X16X4_F32.
// Block = 4 waves computing a 64(M) x 64(N) super-tile. The shared 64x256 B
// tile is staged into LDS once per block (async global->LDS copies when the
// toolchain exposes them), then each wave runs a 64-step K loop:
//   A frag (16x4) from global (own 16 rows), B frags (4x16) from LDS.
// Layouts (wave32): A lane l<16 -> row M=l, K={k,k+1}; lanes 16.. -> K+2.
//                   B lane l<16 -> col N=l; halves carry K / K+2.
//                   D vgpr d: lanes 0-15 M=d, lanes 16-31 M=d+8; N=lane%16.
// ---------------------------------------------------------------------------
__global__ void __launch_bounds__(128)
gemm_kernel(const float* __restrict__ An, const float* __restrict__ Bn,
            float* __restrict__ sim) {
    __shared__ float Bt[64 * BT_STRIDE];

    const int tid    = threadIdx.x;          // 0..127
    const int waveId = tid >> 5;             // 0..3
    const int lane   = tid & 31;
    const int half   = lane >> 4;            // 0 or 1
    const int l16    = lane & 15;
    const int waveM  = blockIdx.y * 64 + waveId * 16;
    const int waveN  = blockIdx.x * 64;

    // ---- stage B tile (64 rows x 256 floats = 64KB) into LDS ----
    {
        const float* gbase = Bn + (size_t)waveN * DIM;
#pragma unroll
        for (int i = 0; i < 32; ++i) {
            int c    = tid + 128 * i;        // 16B chunk id, 0..4095
            int row  = c >> 6;               // 64 chunks per 1KB row
            int col4 = (c & 63) * 4;         // float index within row
            copy16_g2lds(gbase + (size_t)row * DIM + col4,
                         &Bt[row * BT_STRIDE + col4]);
        }
        wait_async_copies();
        __syncthreads();
    }

    const float* arow  = An + (size_t)(waveM + l16) * DIM;
    const float* blane = &Bt[l16 * BT_STRIDE];

    v8f acc0 = {}, acc1 = {}, acc2 = {}, acc3 = {};

#pragma unroll 4
    for (int k = 0; k < DIM; k += 4) {
        const int ko = k + 2 * half;
        v2f a  = *(const v2f*)(arow + ko);
        v2f b0 = *(const v2f*)(blane + 0 * 16 * BT_STRIDE + ko);
        v2f b1 = *(const v2f*)(blane + 1 * 16 * BT_STRIDE + ko);
        v2f b2 = *(const v2f*)(blane + 2 * 16 * BT_STRIDE + ko);
        v2f b3 = *(const v2f*)(blane + 3 * 16 * BT_STRIDE + ko);
        acc0 = __builtin_amdgcn_wmma_f32_16x16x4_f32(false, a, false, b0,
                                                     (short)0, acc0, false, false);
        acc1 = __builtin_amdgcn_wmma_f32_16x16x4_f32(false, a, false, b1,
                                                     (short)0, acc1, false, false);
        acc2 = __builtin_amdgcn_wmma_f32_16x16x4_f32(false, a, false, b2,
                                                     (short)0, acc2, false, false);
        acc3 = __builtin_amdgcn_wmma_f32_16x16x4_f32(false, a, false, b3,
                                                     (short)0, acc3, false, false);
    }

    const size_t col  = (size_t)waveN + l16;
    const int    rowb = waveM + 8 * half;
#pragma unroll
    for (int d = 0; d < 8; ++d) {
        size_t rof = (size_t)(rowb + d) * QUEUE_N;
        sim[rof + col +  0] = acc0[d];
        sim[rof + col + 16] = acc1[d];
        sim[rof + col + 32] = acc2[d];
        sim[rof + col + 48] = acc3[d];
    }
}

// ---------------------------------------------------------------------------
// Phase 2: per-row single-pass reduction (online softmax + per-class sums +
// exact top-5). FUSED=true recomputes sim on the fly (ws-too-small fallback).
// ---------------------------------------------------------------------------
template <bool FUSED>
__global__ void __launch_bounds__(256)
rowloss_kernel(const float* __restrict__ sim, const float* __restrict__ An,
               const float* __restrict__ Bn, const int* __restrict__ labels,
               float* __restrict__ per_sample) {
    __shared__ float clsSum[NCLS];
    __shared__ float clsCnt[NCLS];
    __shared__ float redM[256];
    __shared__ float redS[256];
    __shared__ unsigned long long cand[256 * KTOP];
    __shared__ float arow[DIM];

    const int row = blockIdx.x;
    const int tid = threadIdx.x;

    if (tid < NCLS) { clsSum[tid] = 0.f; clsCnt[tid] = 0.f; }
    if (FUSED) arow[tid] = An[(size_t)row * DIM + tid];
    __syncthreads();

    unsigned long long top[KTOP];
#pragma unroll
    for (int i = 0; i < KTOP; ++i) top[i] = 0ull;   // sorted desc

    float m = -INFINITY, s = 0.f;

    for (int j = tid; j < QUEUE_N; j += 256) {
        float v;
        if (FUSED) {
            const float* brow = Bn + (size_t)j * DIM;
            float acc = 0.f;
#pragma unroll 4
            for (int k = 0; k < DIM; k += 4) {
                v4f b = *(const v4f*)(brow + k);
                acc += arow[k] * b.x + arow[k + 1] * b.y +
                       arow[k + 2] * b.z + arow[k + 3] * b.w;
            }
            v = acc;
        } else {
            v = sim[(size_t)row * QUEUE_N + j];
        }

        float x = v / 0.07f;

        if (x > m) { s = s * __expf(m - x) + 1.f; m = x; }
        else       { s += __expf(x - m); }

        int lab = labels[j];
        atomicAdd(&clsSum[lab], x);
        atomicAdd(&clsCnt[lab], 1.f);

        unsigned int u = __float_as_uint(v);
        u = (u & 0x80000000u) ? ~u : (u | 0x80000000u);
        unsigned long long key =
            ((unsigned long long)u << 32) | (unsigned int)(~(unsigned)j);
        if (key > top[KTOP - 1]) {
            top[KTOP - 1] = key;
#pragma unroll
            for (int i = KTOP - 1; i > 0; --i)
                if (top[i] > top[i - 1]) {
                    unsigned long long t = top[i];
                    top[i] = top[i - 1];
                    top[i - 1] = t;
                }
        }
    }

    redM[tid] = m;
    redS[tid] = s;
#pragma unroll
    for (int i = 0; i < KTOP; ++i) cand[tid * KTOP + i] = top[i];
    __syncthreads();

    for (int off = 128; off >= 1; off >>= 1) {
        if (tid < off) {
            float m1 = redM[tid], s1 = redS[tid];
            float m2 = redM[tid + off], s2 = redS[tid + off];
            float M = fmaxf(m1, m2);
            redM[tid] = M;
            redS[tid] = s1 * __expf(m1 - M) + s2 * __expf(m2 - M);
        }
        __syncthreads();
    }

    if (tid == 0) {
        unsigned long long best[KTOP];
        for (int i = 0; i < KTOP; ++i) {
            unsigned long long bk = 0ull;
            int bi = 0;
            for (int c = 0; c < 256 * KTOP; ++c)
                if (cand[c] > bk) { bk = cand[c]; bi = c; }
            best[i] = bk;
            cand[bi] = 0ull;
        }
        int labs[KTOP];
        for (int i = 0; i < KTOP; ++i) {
            int j = (int)(~(unsigned int)(best[i] & 0xFFFFFFFFull));
            labs[i] = labels[j];
        }
        int bestLab = NCLS, bestCnt = 0;
        for (int i = 0; i < KTOP; ++i) {
            int c = 0;
            for (int q = 0; q < KTOP; ++q) c += (labs[q] == labs[i]);
            if (c > bestCnt || (c == bestCnt && labs[i] < bestLab)) {
                bestCnt = c;
                bestLab = labs[i];
            }
        }
        float cnt  = clsCnt[bestLab];
        float csum = clsSum[bestLab];
        float LSE  = redM[0] + logf(redS[0]);
        per_sample[row] = -(csum - cnt * LSE) / fmaxf(cnt, 1.f);
    }
}

// ---------------------------------------------------------------------------
// Phase 3: mean over 512 rows, apply warm-up coefficient.
// ---------------------------------------------------------------------------
__global__ void __launch_bounds__(256)
finalize_kernel(const float* __restrict__ per_sample,
                const int* __restrict__ it, float* __restrict__ out) {
    __shared__ float red[256];
    int tid = threadIdx.x;
    red[tid] = per_sample[tid] + per_sample[tid + 256];
    __syncthreads();
    for (int off = 128; off >= 1; off >>= 1) {
        if (tid < off) red[tid] += red[tid + off];
        __syncthreads();
    }
    if (tid == 0) {
        float coeff = (it[0] > WARM_UP) ? 0.1f : 0.0f;
        out[0] = coeff * (red[0] / (float)BS);
    }
}

// ---------------------------------------------------------------------------
extern "C" void kernel_launch(void* const* d_in, const int* in_sizes, int n_in,
                              void* d_out, int out_size, void* d_ws,
                              size_t ws_size, hipStream_t stream) {
    (void)in_sizes; (void)n_in; (void)out_size;

    const float* feats = (const float*)d_in[0];   // (1024, 256) f32
    const int*   slab  = (const int*)d_in[1];     // (512,) i32
    const int*   it    = (const int*)d_in[2];     // scalar i32
    const float* queue = (const float*)d_in[3];   // (48000, 256) f32
    const int*   qlab  = (const int*)d_in[4];     // (48000,) i32
    float*       out   = (float*)d_out;

    char*  ws  = (char*)d_ws;
    size_t off = 0;
    auto alloc = [&](size_t bytes) -> void* {
        void* p = ws + off;
        off += (bytes + 255) & ~(size_t)255;
        return p;
    };
    float* An         = (float*)alloc((size_t)BS * DIM * 4);
    float* Bn         = (float*)alloc((size_t)QUEUE_N * DIM * 4);
    int*   labels     = (int*)alloc((size_t)QUEUE_N * 4);
    float* per_sample = (float*)alloc((size_t)BS * 4);
    size_t baseNeed   = off;
    size_t simBytes   = (size_t)BS * QUEUE_N * 4;
    float* sim        = (float*)alloc(simBytes);
    bool   useGemm    = (ws_size >= baseNeed + simBytes);

    // Phase 0: normalize + merge labels (one wave per row, 8 rows/block)
    {
        int waves  = QUEUE_N + BS;
        int blocks = (waves + 7) / 8;
        prep_kernel<<<blocks, 256, 0, stream>>>(feats, slab, queue, qlab,
                                                An, Bn, labels);
    }

    if (useGemm) {
        // Phase 1: fp32 WMMA GEMM, 4 waves/block computing 64x64, LDS-staged B
        dim3 grid(QUEUE_N / 64, BS / 64);   // (750, 8)
        gemm_kernel<<<grid, 128, 0, stream>>>(An, Bn, sim);
        // Phase 2: fused per-row reduction over materialized sim
        rowloss_kernel<false><<<BS, 256, 0, stream>>>(sim, An, Bn, labels,
                                                      per_sample);
    } else {
        // Fallback: recompute sim on the fly (VALU dot products)
        rowloss_kernel<true><<<BS, 256, 0, stream>>>(nullptr, An, Bn, labels,
                                                     per_sample);
    }

    // Phase 3: mean + coefficient
    finalize_kernel<<<1, 256, 0, stream>>>(per_sample, it, out);
}